// ScaledDotProductAttention_28724741276310
// MI455X (gfx1250) — compile-verified
//
#include <hip/hip_runtime.h>
#include <hip/hip_bf16.h>

typedef __attribute__((ext_vector_type(2))) float v2f;
typedef __attribute__((ext_vector_type(4))) float v4f;
typedef __attribute__((ext_vector_type(8))) float v8f;

#define B_ 16
#define L_ 2048
#define D_ 64
// LDS row stride in floats: 2048 + 20 (multiple of 4 -> rows stay 16B aligned
// for b128 access). Stride % 64 == 20 -> consecutive rows land on offset bank
// ranges; 8*stride % 64 == 32 -> WMMA C-fragment stores (rows r and r+8 from
// the two lane halves) hit disjoint banks.
#define ROW_STRIDE 2068
#define NWAVES 8
#define JB_PER_WAVE (128 / NWAVES)   // 16 j-blocks of 16 key columns each

__global__ void __launch_bounds__(256)
attn_softmax_kernel(const float* __restrict__ q,
                    const float* __restrict__ k,
                    const float* __restrict__ v,
                    float* __restrict__ out) {
  extern __shared__ float smem[];            // 16*ROW_STRIDE scores + 16 sums
  float* scores  = smem;
  float* row_inv = smem + 16 * ROW_STRIDE;

  const int tid  = threadIdx.x;
  const int lane = tid & 31;
  const int wave = tid >> 5;
  const int m    = lane & 15;                // tile row (A) / N index (B,C)
  const int half = lane >> 4;

  const int wg = blockIdx.x;                 // 2048 = B * (L/16)
  const int b  = wg >> 7;
  const int qt = wg & 127;

  // ---------- pass-through copy of V (first tuple output, streaming) ----------
  {
    const v4f* v4 = (const v4f*)v;
    v4f*       o4 = (v4f*)out;
    const int  i  = wg * 256 + tid;          // 2048*256*4 = B*L*D floats exactly
    __builtin_nontemporal_store(__builtin_nontemporal_load(v4 + i), o4 + i);
  }

  // ---------- A fragments (q tile), invariant over the j loop ----------
  // 16x4 f32 A layout: lane holds q[m][4*kk + 2*half + {0,1}]
  const float* qrow = q + (size_t)(b * L_ + qt * 16 + m) * D_;
  v2f aq[16];
#pragma unroll
  for (int kk = 0; kk < 16; ++kk)
    aq[kk] = *(const v2f*)(qrow + kk * 4 + 2 * half);

  // ---------- WMMA: scores tile = q_tile @ k_panel^T ----------
  // Two j-blocks per iteration: batched load clauses, then 32 WMMAs as two
  // independent accumulation chains (compiler software-pipelines the body).
  const float* kbase = k + (size_t)b * L_ * D_;
  for (int t = 0; t < JB_PER_WAVE; t += 2) {
    const int jb0 = wave * JB_PER_WAVE + t;
    const float* krow0 = kbase + (size_t)(jb0 * 16 + m) * D_;
    const float* krow1 = krow0 + 16 * D_;
    if (t + 2 < JB_PER_WAVE)
      __builtin_prefetch(krow0 + 32 * D_, 0, 3);  // warm near cache, next pair

    v2f bk0[16], bk1[16];
#pragma unroll
    for (int kk = 0; kk < 16; ++kk)
      bk0[kk] = *(const v2f*)(krow0 + kk * 4 + 2 * half);
#pragma unroll
    for (int kk = 0; kk < 16; ++kk)
      bk1[kk] = *(const v2f*)(krow1 + kk * 4 + 2 * half);

    v8f c0 = {0.f, 0.f, 0.f, 0.f, 0.f, 0.f, 0.f, 0.f};
    v8f c1 = {0.f, 0.f, 0.f, 0.f, 0.f, 0.f, 0.f, 0.f};
#pragma unroll
    for (int kk = 0; kk < 16; ++kk) {
      c0 = __builtin_amdgcn_wmma_f32_16x16x4_f32(false, aq[kk], false, bk0[kk],
                                                 (short)0, c0, false, false);
      c1 = __builtin_amdgcn_wmma_f32_16x16x4_f32(false, aq[kk], false, bk1[kk],
                                                 (short)0, c1, false, false);
    }

    // C layout: VGPR vv -> row (vv + 8*half), column = m
#pragma unroll
    for (int vv = 0; vv < 8; ++vv) {
      scores[(vv + 8 * half) * ROW_STRIDE + jb0 * 16 + m]        = c0[vv];
      scores[(vv + 8 * half) * ROW_STRIDE + (jb0 + 1) * 16 + m]  = c1[vv];
    }
  }
  __syncthreads();

  // ---------- softmax over each of the 16 rows (16 threads per row, b128) ----
  {
    const int r = tid >> 4;
    const int g = tid & 15;
    v4f* rowv = (v4f*)(scores + r * ROW_STRIDE);   // 512 v4f per row

    float mx = -3.402823466e38f;
#pragma unroll 4
    for (int cidx = 0; cidx < 32; ++cidx) {
      v4f x = rowv[g + 16 * cidx];
      mx = fmaxf(mx, fmaxf(fmaxf(x.x, x.y), fmaxf(x.z, x.w)));
    }
    mx = fmaxf(mx, __shfl_xor(mx, 1, 16));
    mx = fmaxf(mx, __shfl_xor(mx, 2, 16));
    mx = fmaxf(mx, __shfl_xor(mx, 4, 16));
    mx = fmaxf(mx, __shfl_xor(mx, 8, 16));

    float s = 0.f;
#pragma unroll 4
    for (int cidx = 0; cidx < 32; ++cidx) {
      v4f x = rowv[g + 16 * cidx];
      v4f e;
      e.x = __expf(x.x - mx);
      e.y = __expf(x.y - mx);
      e.z = __expf(x.z - mx);
      e.w = __expf(x.w - mx);
      rowv[g + 16 * cidx] = e;
      s += (e.x + e.y) + (e.z + e.w);
    }
    s += __shfl_xor(s, 1, 16);
    s += __shfl_xor(s, 2, 16);
    s += __shfl_xor(s, 4, 16);
    s += __shfl_xor(s, 8, 16);
    if (g == 0) row_inv[r] = 1.0f / s;
  }
  __syncthreads();

  // ---------- coalesced, non-temporal b128 write-out of the attention tile ----
  float* attn = out + (size_t)B_ * L_ * D_;
#pragma unroll 1
  for (int r = 0; r < 16; ++r) {
    const float inv  = row_inv[r];
    v4f* orow = (v4f*)(attn + ((size_t)b * L_ + (size_t)(qt * 16 + r)) * L_);
    const v4f* srow = (const v4f*)(scores + r * ROW_STRIDE);
#pragma unroll
    for (int it = 0; it < 2; ++it) {           // 512 v4f per row / 256 threads
      const int j = tid + it * 256;
      v4f x = srow[j];
      v4f y = {x.x * inv, x.y * inv, x.z * inv, x.w * inv};
      __builtin_nontemporal_store(y, orow + j);
    }
  }
}

extern "C" void kernel_launch(void* const* d_in, const int* in_sizes, int n_in,
                              void* d_out, int out_size, void* d_ws, size_t ws_size,
                              hipStream_t stream) {
  (void)in_sizes; (void)n_in; (void)d_ws; (void)ws_size; (void)out_size;
  const float* q = (const float*)d_in[0];
  const float* k = (const float*)d_in[1];
  const float* v = (const float*)d_in[2];
  float* out = (float*)d_out;

  const size_t smem = (size_t)(16 * ROW_STRIDE + 16) * sizeof(float);  // ~132 KB
  hipFuncSetAttribute((const void*)attn_softmax_kernel,
                      hipFuncAttributeMaxDynamicSharedMemorySize, (int)smem);
  attn_softmax_kernel<<<dim3(B_ * (L_ / 16)), dim3(256), smem, stream>>>(q, k, v, out);
}